// GATGraphLevelBinary_43576738185828
// MI455X (gfx1250) — compile-verified
//
#include <hip/hip_runtime.h>
#include <hip/hip_bf16.h>

typedef float v2f __attribute__((ext_vector_type(2)));
typedef float v8f __attribute__((ext_vector_type(8)));

#define N_NODES   50000
#define N_FEAT    128
#define NEDGES    1600000
#define N_GRAPHS  512
#define E_TOT     (NEDGES + N_NODES)      // edges + self loops
#define M_TILES   (N_NODES / 16)          // 3125, exact

// ---------------------------------------------------------------- utilities
__global__ void fill_f32(float* __restrict__ p, int n, float v) {
    int i = blockIdx.x * blockDim.x + threadIdx.x;
    if (i < n) p[i] = v;
}

// float atomic max via signed-max / unsigned-min trick (native L2 atomics)
__device__ __forceinline__ void atomicMaxF32(float* addr, float val) {
    if (val >= 0.0f) {
        atomicMax((int*)addr, __float_as_int(val));
    } else {
        atomicMin((unsigned int*)addr, __float_as_uint(val));
    }
}

__device__ __forceinline__ float lrelu(float x) { return x > 0.0f ? x : 0.2f * x; }

// ---------------------------------------------------------------- WMMA GEMM
// out[M, NCOL] = A[M, K] @ W[K, NCOL], f32, exact. One 16-row tile per wave,
// NCOL/16 accumulators, W staged in LDS. blockDim must be 256 (8 waves).
template <int K, int NCOL>
__global__ void gemm_wmma_f32(const float* __restrict__ A,
                              const float* __restrict__ W,
                              float* __restrict__ out, int m_tiles) {
    __shared__ float wl[K * NCOL];
    for (int i = threadIdx.x; i < K * NCOL; i += blockDim.x) wl[i] = W[i];
    __syncthreads();

    const int wave = threadIdx.x >> 5;
    const int lane = threadIdx.x & 31;
    const int tile = blockIdx.x * (blockDim.x >> 5) + wave;
    if (tile >= m_tiles) return;              // wave-uniform: EXEC stays all-ones

    const int col = lane & 15;                // N index inside 16-wide tile
    const int kb  = (lane >> 4) << 1;         // lanes 16..31 hold K+2,K+3
    constexpr int NT = NCOL / 16;

    v8f acc[NT] = {};
    const float* arow = A + (size_t)(tile * 16 + col) * K;   // row = tile*16 + (lane&15)

    for (int k0 = 0; k0 < K; k0 += 4) {
        v2f a = { arow[k0 + kb], arow[k0 + kb + 1] };
#pragma unroll
        for (int t = 0; t < NT; ++t) {
            v2f b = { wl[(k0 + kb)     * NCOL + t * 16 + col],
                      wl[(k0 + kb + 1) * NCOL + t * 16 + col] };
            acc[t] = __builtin_amdgcn_wmma_f32_16x16x4_f32(
                false, a, false, b, (short)0, acc[t], false, false);
        }
    }

    const int rbase = tile * 16 + ((lane >> 4) << 3);   // M = v + 8*(lane/16)
#pragma unroll
    for (int t = 0; t < NT; ++t)
#pragma unroll
        for (int v = 0; v < 8; ++v)
            out[(size_t)(rbase + v) * NCOL + t * 16 + col] = acc[t][v];
}

// ---------------------------------------------------------------- attention logits
// layer1: H=8, C=8.  as/ad[n*8+h] = sum_c h1[n,64] segment * att[h,8]
__global__ void alpha8_kernel(const float* __restrict__ h,
                              const float* __restrict__ att_s,
                              const float* __restrict__ att_d,
                              float* __restrict__ as_, float* __restrict__ ad_) {
    int tid = blockIdx.x * blockDim.x + threadIdx.x;
    if (tid >= N_NODES * 8) return;
    int n = tid >> 3, hd = tid & 7;
    const float* row = h + (size_t)n * 64 + hd * 8;
    float s = 0.f, d = 0.f;
#pragma unroll
    for (int c = 0; c < 8; ++c) {
        float v = row[c];
        s += v * att_s[hd * 8 + c];
        d += v * att_d[hd * 8 + c];
    }
    as_[tid] = s; ad_[tid] = d;
}

// layer2: H=1, C=16
__global__ void alpha16_kernel(const float* __restrict__ h,
                               const float* __restrict__ att_s,
                               const float* __restrict__ att_d,
                               float* __restrict__ as_, float* __restrict__ ad_) {
    int n = blockIdx.x * blockDim.x + threadIdx.x;
    if (n >= N_NODES) return;
    const float* row = h + (size_t)n * 16;
    float s = 0.f, d = 0.f;
#pragma unroll
    for (int c = 0; c < 16; ++c) {
        float v = row[c];
        s += v * att_s[c];
        d += v * att_d[c];
    }
    as_[n] = s; ad_[n] = d;
}

// ---------------------------------------------------------------- edge passes
__device__ __forceinline__ void edge_endpoints(const int* __restrict__ ei, int e,
                                               int& s, int& d) {
    if (e < NEDGES) { s = ei[e]; d = ei[NEDGES + e]; }
    else            { s = d = e - NEDGES; }           // self loop
}

// layer1 pass A: segment max of leaky-relu logits, thread per (edge, head)
__global__ void edge_max8(const int* __restrict__ ei,
                          const float* __restrict__ as_, const float* __restrict__ ad_,
                          float* __restrict__ m) {
    int tid = blockIdx.x * blockDim.x + threadIdx.x;
    if (tid >= E_TOT * 8) return;
    int e = tid >> 3, h = tid & 7;
    int s, d; edge_endpoints(ei, e, s, d);
    float v = lrelu(as_[s * 8 + h] + ad_[d * 8 + h]);
    atomicMaxF32(&m[d * 8 + h], v);
}

// layer1 pass B: p = exp(e - m[dst]); agg[dst] += p*h[src]; denom[dst] += p
// thread per (edge, feature j of 64); h = j>>3
__global__ void edge_agg8(const int* __restrict__ ei,
                          const float* __restrict__ as_, const float* __restrict__ ad_,
                          const float* __restrict__ m, const float* __restrict__ h1,
                          float* __restrict__ agg, float* __restrict__ denom) {
    int tid = blockIdx.x * blockDim.x + threadIdx.x;
    if (tid >= E_TOT * 64) return;
    int e = tid >> 6, j = tid & 63, h = j >> 3;
    int s, d; edge_endpoints(ei, e, s, d);
    float v = lrelu(as_[s * 8 + h] + ad_[d * 8 + h]);
    float p = __expf(v - m[d * 8 + h]);
    atomicAdd(&agg[(size_t)d * 64 + j], p * h1[(size_t)s * 64 + j]);
    if ((j & 7) == 0) atomicAdd(&denom[d * 8 + h], p);
}

// layer1 epilogue: normalize, +bias, ELU; write in place over h1
__global__ void node_epi1(const float* __restrict__ agg, const float* __restrict__ denom,
                          const float* __restrict__ b1, float* __restrict__ h1) {
    int tid = blockIdx.x * blockDim.x + threadIdx.x;
    if (tid >= N_NODES * 64) return;
    int n = tid >> 6, j = tid & 63, h = j >> 3;
    float v = agg[tid] / denom[n * 8 + h] + b1[j];
    h1[tid] = v > 0.f ? v : __expf(v) - 1.f;     // ELU
}

// layer2 pass A: thread per edge (H=1)
__global__ void edge_max1(const int* __restrict__ ei,
                          const float* __restrict__ as_, const float* __restrict__ ad_,
                          float* __restrict__ m) {
    int e = blockIdx.x * blockDim.x + threadIdx.x;
    if (e >= E_TOT) return;
    int s, d; edge_endpoints(ei, e, s, d);
    atomicMaxF32(&m[d], lrelu(as_[s] + ad_[d]));
}

// layer2 pass B: thread per (edge, feature c of 16)
__global__ void edge_agg1(const int* __restrict__ ei,
                          const float* __restrict__ as_, const float* __restrict__ ad_,
                          const float* __restrict__ m, const float* __restrict__ h2,
                          float* __restrict__ agg, float* __restrict__ denom) {
    int tid = blockIdx.x * blockDim.x + threadIdx.x;
    if (tid >= E_TOT * 16) return;
    int e = tid >> 4, c = tid & 15;
    int s, d; edge_endpoints(ei, e, s, d);
    float v = lrelu(as_[s] + ad_[d]);
    float p = __expf(v - m[d]);
    atomicAdd(&agg[(size_t)d * 16 + c], p * h2[(size_t)s * 16 + c]);
    if (c == 0) atomicAdd(&denom[d], p);
}

// layer2 epilogue fused with graph mean-pool accumulation
__global__ void node_epi2_pool(const float* __restrict__ agg, const float* __restrict__ denom,
                               const float* __restrict__ b2, const int* __restrict__ batch,
                               float* __restrict__ gsum, float* __restrict__ gcnt) {
    int tid = blockIdx.x * blockDim.x + threadIdx.x;
    if (tid >= N_NODES * 16) return;
    int n = tid >> 4, c = tid & 15;
    float v = agg[tid] / denom[n] + b2[c];
    int g = batch[n];
    atomicAdd(&gsum[g * 16 + c], v);
    if (c == 0) atomicAdd(&gcnt[g], 1.0f);
}

// final: per-graph mean, linear(16->1), sigmoid
__global__ void final_head(const float* __restrict__ gsum, const float* __restrict__ gcnt,
                           const float* __restrict__ lw, const float* __restrict__ lb,
                           float* __restrict__ out) {
    int g = blockIdx.x * blockDim.x + threadIdx.x;
    if (g >= N_GRAPHS) return;
    float inv = 1.0f / fmaxf(gcnt[g], 1.0f);
    float acc = lb[0];
#pragma unroll
    for (int c = 0; c < 16; ++c) acc += gsum[g * 16 + c] * inv * lw[c];
    out[g] = 1.0f / (1.0f + __expf(-acc));
}

// ---------------------------------------------------------------- launcher
extern "C" void kernel_launch(void* const* d_in, const int* in_sizes, int n_in,
                              void* d_out, int out_size, void* d_ws, size_t ws_size,
                              hipStream_t stream) {
    const float* x    = (const float*)d_in[0];
    const int*   ei   = (const int*)d_in[1];
    const int*   batch= (const int*)d_in[2];
    const float* W1   = (const float*)d_in[3];
    const float* aS1  = (const float*)d_in[4];
    const float* aD1  = (const float*)d_in[5];
    const float* b1   = (const float*)d_in[6];
    const float* W2   = (const float*)d_in[7];
    const float* aS2  = (const float*)d_in[8];
    const float* aD2  = (const float*)d_in[9];
    const float* b2   = (const float*)d_in[10];
    const float* lw   = (const float*)d_in[11];
    const float* lb   = (const float*)d_in[12];
    float* out = (float*)d_out;

    // ---- workspace layout (floats). Zero-initialized region is contiguous.
    float* ws = (float*)d_ws;
    size_t o = 0;
    float* h1     = ws + o; o += (size_t)N_NODES * 64;   // gemm1 out, later elu'd
    float* agg1   = ws + o; o += (size_t)N_NODES * 64;   // zero region starts here
    float* denom1 = ws + o; o += (size_t)N_NODES * 8;
    float* agg2   = ws + o; o += (size_t)N_NODES * 16;
    float* denom2 = ws + o; o += (size_t)N_NODES;
    float* gsum   = ws + o; o += (size_t)N_GRAPHS * 16;
    float* gcnt   = ws + o; o += (size_t)N_GRAPHS;       // zero region ends here
    int zero_cnt  = (int)((ws + o) - agg1);
    float* m1     = ws + o; o += (size_t)N_NODES * 8;    // -inf region (m1+m2)
    float* m2     = ws + o; o += (size_t)N_NODES;
    int minf_cnt  = N_NODES * 9;
    float* as1    = ws + o; o += (size_t)N_NODES * 8;
    float* ad1    = ws + o; o += (size_t)N_NODES * 8;
    float* as2    = ws + o; o += (size_t)N_NODES;
    float* ad2    = ws + o; o += (size_t)N_NODES;
    float* h2a    = ws + o; o += (size_t)N_NODES * 16;   // gemm2 out

    const int T = 256;
    auto nb = [](long long n, int t) { return (unsigned)((n + t - 1) / t); };

    // 1) init accumulators
    fill_f32<<<nb(zero_cnt, T), T, 0, stream>>>(agg1, zero_cnt, 0.0f);
    fill_f32<<<nb(minf_cnt, T), T, 0, stream>>>(m1, minf_cnt, -3.0e38f);

    // 2) layer 1: h1 = x @ W1  (f32 WMMA)
    gemm_wmma_f32<128, 64><<<nb(M_TILES, 8), 256, 0, stream>>>(x, W1, h1, M_TILES);
    // 3) attention logits
    alpha8_kernel<<<nb((long long)N_NODES * 8, T), T, 0, stream>>>(h1, aS1, aD1, as1, ad1);
    // 4) segment max
    edge_max8<<<nb((long long)E_TOT * 8, T), T, 0, stream>>>(ei, as1, ad1, m1);
    // 5) exp + weighted scatter + denom
    edge_agg8<<<nb((long long)E_TOT * 64, T), T, 0, stream>>>(ei, as1, ad1, m1, h1, agg1, denom1);
    // 6) normalize + bias + ELU (in place into h1)
    node_epi1<<<nb((long long)N_NODES * 64, T), T, 0, stream>>>(agg1, denom1, b1, h1);

    // 7) layer 2: h2a = h1 @ W2 (f32 WMMA)
    gemm_wmma_f32<64, 16><<<nb(M_TILES, 8), 256, 0, stream>>>(h1, W2, h2a, M_TILES);
    // 8) attention logits
    alpha16_kernel<<<nb(N_NODES, T), T, 0, stream>>>(h2a, aS2, aD2, as2, ad2);
    // 9) segment max
    edge_max1<<<nb(E_TOT, T), T, 0, stream>>>(ei, as2, ad2, m2);
    // 10) exp + weighted scatter + denom
    edge_agg1<<<nb((long long)E_TOT * 16, T), T, 0, stream>>>(ei, as2, ad2, m2, h2a, agg2, denom2);
    // 11) normalize + bias + mean-pool accumulate
    node_epi2_pool<<<nb((long long)N_NODES * 16, T), T, 0, stream>>>(agg2, denom2, b2, batch, gsum, gcnt);
    // 12) linear + sigmoid
    final_head<<<nb(N_GRAPHS, T), T, 0, stream>>>(gsum, gcnt, lw, lb, out);
}